// RegularizationLoss_69114613729479
// MI455X (gfx1250) — compile-verified
//
#include <hip/hip_runtime.h>

typedef __attribute__((ext_vector_type(2))) float v2f;
typedef __attribute__((ext_vector_type(4))) float v4f;
typedef __attribute__((ext_vector_type(8))) float v8f;

// Exact fp32 sum of one wave32's lane values using V_WMMA_F32_16X16X4_F32.
// A (16x4, striped): VGPR0 = x, VGPR1 = 0  ->  row i = {x_i, 0, x_{i+16}, 0}
// B (4x16) = all ones  ->  D_{i,j} = x_i + x_{i+16} for every column j.
// D layout: lanes 0-15 VGPR r hold M=r; lanes 16-31 hold M=8+r.
// So sum(D[0..7]) on lane 0 = sum over M=0..7, on lane 16 = sum over M=8..15.
__device__ __forceinline__ float wave32_sum_wmma(float x) {
    v2f a; a[0] = x;    a[1] = 0.0f;
    v2f b; b[0] = 1.0f; b[1] = 1.0f;
    v8f c = {};
    v8f d = __builtin_amdgcn_wmma_f32_16x16x4_f32(
        /*neg_a=*/false, a, /*neg_b=*/false, b,
        /*c_mod=*/(short)0, c, /*reuse_a=*/false, /*reuse_b=*/false);
    float s = ((d[0] + d[1]) + (d[2] + d[3])) + ((d[4] + d[5]) + (d[6] + d[7]));
    // total = (partial over M=0..7) + (partial over M=8..15)
    return __shfl(s, 0, 32) + __shfl(s, 16, 32);
}

__device__ __forceinline__ float dot3(const float* p, const float* q) {
    return p[0] * q[0] + p[1] * q[1] + p[2] * q[2];
}

// Pass 1: streaming read of (B,6,4) fp32, per-thread loss accumulation,
// WMMA wave reduce + LDS block reduce, one partial per block.
__global__ void reg_loss_partial(const float* __restrict__ in,
                                 float* __restrict__ ws, int B) {
    const v4f* in4 = (const v4f*)in;
    float acc = 0.0f;
    const int stride = gridDim.x * blockDim.x;
    for (int bidx = blockIdx.x * blockDim.x + threadIdx.x; bidx < B; bidx += stride) {
        const v4f* p = in4 + (size_t)bidx * 6;
        v4f r[6];
#pragma unroll
        for (int i = 0; i < 6; ++i)
            r[i] = __builtin_nontemporal_load(p + i);   // global_load_b128, NT (stream past L2)

        float u[6][3];
#pragma unroll
        for (int i = 0; i < 6; ++i) {
            float x = r[i][1], y = r[i][2], z = r[i][3];   // columns 1:4
            float inv = rsqrtf(x * x + y * y + z * z);
            u[i][0] = x * inv; u[i][1] = y * inv; u[i][2] = z * inv;
        }
        // G = M M^T - I is symmetric: loss = sum(diag^2) + 2*sum(offdiag^2)
#pragma unroll
        for (int k = 0; k < 2; ++k) {
            const float* a = u[3 * k + 0];
            const float* b = u[3 * k + 1];
            const float* c = u[3 * k + 2];
            float daa = dot3(a, a) - 1.0f;
            float dbb = dot3(b, b) - 1.0f;
            float dcc = dot3(c, c) - 1.0f;
            float cab = dot3(a, b), cac = dot3(a, c), cbc = dot3(b, c);
            acc += (daa * daa + dbb * dbb + dcc * dcc)
                 + 2.0f * (cab * cab + cac * cac + cbc * cbc);
        }
    }

    // Wave reduce (EXEC all-ones here: loop has reconverged, block fully populated)
    float wsum = wave32_sum_wmma(acc);

    __shared__ float warp_sums[8];
    const int lane = threadIdx.x & 31;
    const int wid  = threadIdx.x >> 5;
    if (lane == 0) warp_sums[wid] = wsum;
    __syncthreads();
    if (wid == 0) {
        float t = (lane < 8) ? warp_sums[lane] : 0.0f;
        t += __shfl_down(t, 4, 32);
        t += __shfl_down(t, 2, 32);
        t += __shfl_down(t, 1, 32);
        if (lane == 0) ws[blockIdx.x] = t;
    }
}

// Pass 2: single block folds the per-block partials, writes mean (fresh store,
// no accumulation -> deterministic across graph replays).
__global__ void reg_loss_final(const float* __restrict__ ws,
                               float* __restrict__ out, int n, float invB) {
    float acc = 0.0f;
    for (int i = threadIdx.x; i < n; i += blockDim.x) acc += ws[i];

    float wsum = wave32_sum_wmma(acc);

    __shared__ float warp_sums[8];
    const int lane = threadIdx.x & 31;
    const int wid  = threadIdx.x >> 5;
    if (lane == 0) warp_sums[wid] = wsum;
    __syncthreads();
    if (wid == 0) {
        float t = (lane < 8) ? warp_sums[lane] : 0.0f;
        t += __shfl_down(t, 4, 32);
        t += __shfl_down(t, 2, 32);
        t += __shfl_down(t, 1, 32);
        if (lane == 0) out[0] = t * invB;
    }
}

extern "C" void kernel_launch(void* const* d_in, const int* in_sizes, int n_in,
                              void* d_out, int out_size, void* d_ws, size_t ws_size,
                              hipStream_t stream) {
    const float* in = (const float*)d_in[0];
    float* out = (float*)d_out;
    float* ws  = (float*)d_ws;

    const int B = in_sizes[0] / 24;   // (B, 6, 4) fp32

    int NB = 4096;                    // 4096 blocks x 256 thr = 1M threads, 4 elems/thread
    size_t max_parts = ws_size / sizeof(float);
    if ((size_t)NB > max_parts) NB = (int)max_parts;
    if (NB < 1) NB = 1;
    const int NT = 256;

    reg_loss_partial<<<NB, NT, 0, stream>>>(in, ws, B);
    reg_loss_final<<<1, 256, 0, stream>>>(ws, out, NB, 1.0f / (float)B);
}